// RiemannianDistanceLoss_67946382623432
// MI455X (gfx1250) — compile-verified
//
#include <hip/hip_runtime.h>
#include <math.h>

// ---------------------------------------------------------------------------
// Riemannian (affine-invariant) SPD distance, batched 32x32, fp32 WMMA path.
//   d(x,y) = || logm( y^{-1/2} x y^{-1/2} ) ||_F ,  output = mean over batch
// Matmul-only: Newton-Schulz invsqrt/sqrt + Mercator log series, all 32x32
// GEMMs on v_wmma_f32_16x16x4_f32. Elementwise work fused into WMMA epilogues;
// all normalization scales folded into scalars (NS is scale-invariant).
// ---------------------------------------------------------------------------

typedef __attribute__((ext_vector_type(2))) float v2f;
typedef __attribute__((ext_vector_type(8))) float v8f;

#define NS_ITERS  12   // Newton-Schulz iterations per (inv)sqrt
#define LOG_TERMS 16   // Mercator series terms for log(I+E), ||E|| <= 0.5

// Wave-local LDS ordering: lanes run in lockstep; drain the LDS unit and stop
// compiler reordering across the point.
__device__ __forceinline__ void wave_lds_sync() {
  __builtin_amdgcn_wave_barrier();
  asm volatile("s_wait_dscnt 0" ::: "memory");
  __builtin_amdgcn_wave_barrier();
}

// Epilogue modes for the fused 32x32 GEMM.
#define MM_PLAIN 0  // C = A*B
#define MM_NS3I  1  // C = 0.5*(3I - A*B)      (Newton-Schulz step)
#define MM_SER   2  // C = A*B and L += coef*C (log-series accumulation)

// C = f(A * B) for 32x32 row-major f32 matrices in LDS. One wave.
// 2x2 tiles of 16x16, K consumed 4 at a time -> 32 x v_wmma_f32_16x16x4_f32.
template <int MODE>
__device__ __forceinline__ void mm32(const float* A, const float* B, float* C,
                                     int lane, float* Lacc = nullptr,
                                     float coef = 0.0f) {
  const int half = lane >> 4;   // 0: lanes 0-15, 1: lanes 16-31
  const int l15  = lane & 15;
  v8f acc00 = {}, acc01 = {}, acc10 = {}, acc11 = {};
#pragma unroll
  for (int k0 = 0; k0 < 32; k0 += 4) {
    const int kb = k0 + (half << 1);  // this lane-half holds K = kb, kb+1
    // A fragments (16x4): lane -> M row, VGPR0/1 -> kb/kb+1
    v2f a0 = {A[l15 * 32 + kb],        A[l15 * 32 + kb + 1]};
    v2f a1 = {A[(16 + l15) * 32 + kb], A[(16 + l15) * 32 + kb + 1]};
    // B fragments (4x16): lane -> N col, VGPR0/1 -> kb/kb+1
    v2f b0 = {B[kb * 32 + l15],        B[(kb + 1) * 32 + l15]};
    v2f b1 = {B[kb * 32 + 16 + l15],   B[(kb + 1) * 32 + 16 + l15]};
    acc00 = __builtin_amdgcn_wmma_f32_16x16x4_f32(false, a0, false, b0,
                                                  (short)0, acc00, false, false);
    acc01 = __builtin_amdgcn_wmma_f32_16x16x4_f32(false, a0, false, b1,
                                                  (short)0, acc01, false, false);
    acc10 = __builtin_amdgcn_wmma_f32_16x16x4_f32(false, a1, false, b0,
                                                  (short)0, acc10, false, false);
    acc11 = __builtin_amdgcn_wmma_f32_16x16x4_f32(false, a1, false, b1,
                                                  (short)0, acc11, false, false);
  }
  // C/D layout: VGPR r holds row (8*half + r) of its 16x16 tile, col = l15.
#pragma unroll
  for (int r = 0; r < 8; ++r) {
    const int m    = half * 8 + r;       // row within tile
    const bool dia = (m == l15);         // diagonal only in (0,0)/(1,1) tiles
    float v00 = acc00[r], v01 = acc01[r], v10 = acc10[r], v11 = acc11[r];
    if (MODE == MM_NS3I) {
      v00 = 0.5f * ((dia ? 3.0f : 0.0f) - v00);
      v11 = 0.5f * ((dia ? 3.0f : 0.0f) - v11);
      v01 = -0.5f * v01;
      v10 = -0.5f * v10;
    }
    if (MODE == MM_SER) {
      Lacc[r]      += coef * v00;
      Lacc[8 + r]  += coef * v10;
      Lacc[16 + r] += coef * v01;
      Lacc[24 + r] += coef * v11;
    }
    C[m * 32 + l15]             = v00;
    C[m * 32 + 16 + l15]        = v01;
    C[(16 + m) * 32 + l15]      = v10;
    C[(16 + m) * 32 + 16 + l15] = v11;
  }
  wave_lds_sync();
}

// Frobenius norm squared; every lane returns the full sum.
__device__ __forceinline__ float frob2(const float* A, int lane) {
  float s = 0.f;
#pragma unroll
  for (int j = 0; j < 32; ++j) {
    float v = A[j * 32 + lane];
    s += v * v;
  }
#pragma unroll
  for (int m = 16; m >= 1; m >>= 1) s += __shfl_xor(s, m, 32);
  return s;
}

struct NSOut {
  float* Y;   // ~ (A/c)^{ 1/2}
  float* Z;   // ~ (A/c)^{-1/2}
  float* w1;  // leftover scratch buffer
  float* w2;  // leftover scratch buffer
  float  c;   // ||A||_F
};

// Coupled Newton-Schulz on A/||A||_F (spectrum in (0,1] for SPD A):
//   T = (3I - Z*Y)/2 ; Y <- Y*T ; Z <- T*Z
// Scale-invariant in A up to the returned c.
__device__ __forceinline__ NSOut ns_pair(const float* A, float* b0, float* b1,
                                         float* b2, float* b3, int lane) {
  const float c  = sqrtf(frob2(A, lane));
  const float rc = 1.0f / c;
#pragma unroll
  for (int j = 0; j < 32; ++j) {
    b0[j * 32 + lane] = A[j * 32 + lane] * rc;       // Y0 = A/c
    b1[j * 32 + lane] = (j == lane) ? 1.0f : 0.0f;   // Z0 = I
  }
  wave_lds_sync();
  float* Y = b0;
  float* Z = b1;
  float* T = b2;
  float* S = b3;
  for (int it = 0; it < NS_ITERS; ++it) {
    mm32<MM_NS3I>(Z, Y, T, lane);   // T = (3I - Z*Y)/2, fused epilogue
    mm32<MM_PLAIN>(Y, T, S, lane);  // Ynew = Y*T
    mm32<MM_PLAIN>(T, Z, Y, lane);  // Znew = T*Z (old Y buffer now free)
    float* spare = Z;
    Z = Y;   // Znew lives in old Y buffer
    Y = S;   // Ynew
    S = spare;
  }
  NSOut o;
  o.Y = Y; o.Z = Z; o.w1 = S; o.w2 = T; o.c = c;
  return o;
}

__global__ __launch_bounds__(64) void riemann_dist_kernel(
    const float* __restrict__ x, const float* __restrict__ y,
    float* __restrict__ out, int Bc, float invB) {
  __shared__ __align__(16) float lds[2][6][1024];
  const int wave = threadIdx.x >> 5;
  const int lane = threadIdx.x & 31;
  const int mat  = blockIdx.x * 2 + wave;
  if (mat >= Bc) return;   // wave-uniform: EXEC stays all-ones for WMMA

  float* b[6];
#pragma unroll
  for (int i = 0; i < 6; ++i) b[i] = &lds[wave][i][0];

  // ---- load x -> b[0], y -> b[1] (coalesced 16B per lane) ----
  const float4* gx = (const float4*)(x + (size_t)mat * 1024);
  const float4* gy = (const float4*)(y + (size_t)mat * 1024);
  float4* lx = (float4*)b[0];
  float4* ly = (float4*)b[1];
  for (int i = lane; i < 256; i += 32) {
    lx[i] = gx[i];
    ly[i] = gy[i];
  }
  wave_lds_sync();

  // ---- invsqrt of y (up to scale): Zy = sqrt(cy) * y^{-1/2} ----
  NSOut ny = ns_pair(b[1], b[2], b[3], b[4], b[5], lane);

  // ---- M' = Zy * x * Zy = cy * M  (NS is scale-invariant; fold cy later) ----
  mm32<MM_PLAIN>(ny.Z, b[0], b[1], lane);   // b[1] = Zy*x  (raw y buffer free)
  mm32<MM_PLAIN>(b[1], ny.Z, ny.w1, lane);  // M' in ny.w1
  float* Mp = ny.w1;

  // ---- two matrix square roots, unscaled; compose scales in scalars ----
  NSOut s1 = ns_pair(Mp, b[1], ny.Y, ny.Z, ny.w2, lane);    // Y1 ~ (M'/c1m)^1/2
  NSOut s2 = ns_pair(s1.Y, Mp, s1.Z, s1.w1, s1.w2, lane);   // Y2 ~ (Y1/c2m)^1/2
  // True M = M'/cy ; c1 = c1m/cy ; M^{1/4} = c1^{1/4} * sqrt(c2m) * Y2
  const float c1    = s1.c / ny.c;
  const float sigma = sqrtf(sqrtf(c1)) * sqrtf(s2.c);

  // ---- E = sigma*Y2 - I (in place) and L init, in epilogue ownership ----
  // Lane owns rows {8h..8h+7, 16+8h..16+8h+7} x cols {l15, 16+l15}.
  const int half = lane >> 4;
  const int l15  = lane & 15;
  float* E = s2.Y;
  float L[32];
#pragma unroll
  for (int r = 0; r < 8; ++r) {
    const int m    = half * 8 + r;
    const bool dia = (m == l15);
    float e00 = sigma * E[m * 32 + l15]             - (dia ? 1.0f : 0.0f);
    float e01 = sigma * E[m * 32 + 16 + l15];
    float e10 = sigma * E[(16 + m) * 32 + l15];
    float e11 = sigma * E[(16 + m) * 32 + 16 + l15] - (dia ? 1.0f : 0.0f);
    E[m * 32 + l15]             = e00;
    E[m * 32 + 16 + l15]        = e01;
    E[(16 + m) * 32 + l15]      = e10;
    E[(16 + m) * 32 + 16 + l15] = e11;
    L[r]      = e00;   // series term i = 1
    L[8 + r]  = e10;
    L[16 + r] = e01;
    L[24 + r] = e11;
  }
  wave_lds_sync();

  // ---- L = log(I+E) = sum (-1)^{i+1} E^i / i, accumulated in registers ----
  float* P   = E;
  float* tgt = s2.w1;   // ping-pong power buffers, never aliasing E
  for (int i = 2; i <= LOG_TERMS; ++i) {
    const float coef = ((i & 1) ? 1.0f : -1.0f) / (float)i;
    mm32<MM_SER>(P, E, tgt, lane, L, coef);          // tgt = E^i, L += coef*tgt
    P   = tgt;
    tgt = (tgt == s2.w1) ? s2.w2 : s2.w1;
  }

  float s = 0.f;
#pragma unroll
  for (int j = 0; j < 32; ++j) s += L[j] * L[j];
#pragma unroll
  for (int m = 16; m >= 1; m >>= 1) s += __shfl_xor(s, m, 32);

  const float d = 4.0f * sqrtf(s);                   // 2^k with k = 2 sqrts
  if (lane == 0) atomicAdd(out, d * invB);
}

extern "C" void kernel_launch(void* const* d_in, const int* in_sizes, int n_in,
                              void* d_out, int out_size, void* d_ws,
                              size_t ws_size, hipStream_t stream) {
  const float* x = (const float*)d_in[0];
  const float* y = (const float*)d_in[1];
  float* out = (float*)d_out;
  const int Bc = in_sizes[0] / 1024;                 // 32*32 per matrix

  hipMemsetAsync(out, 0, (size_t)out_size * sizeof(float), stream);

  const int blocks = (Bc + 1) / 2;                   // 2 waves (matrices)/block
  riemann_dist_kernel<<<blocks, 64, 0, stream>>>(x, y, out, Bc,
                                                 1.0f / (float)Bc);
}